// DSLayer_36283883716971
// MI455X (gfx1250) — compile-verified
//
#include <hip/hip_runtime.h>

#define N_NODES 50000
#define N_EDGES 400000
#define IN_F    128
#define POS_F   64
#define EMB_F   128
#define NTILES  (N_NODES / 16)      // 3125
#define WPB     4                   // waves per block
#define LNE     1e-5f

typedef float v2f __attribute__((ext_vector_type(2)));
typedef float v4f __attribute__((ext_vector_type(4)));
typedef float v8f __attribute__((ext_vector_type(8)));

// ---------------------------------------------------------------------------
// Edge scatter: agg_x[d] += x[s], agg_pos[d] += pos[s]  (one edge per block)
// ---------------------------------------------------------------------------
__global__ __launch_bounds__(192) void scatter_kernel(
    const float* __restrict__ x, const float* __restrict__ pos,
    const int* __restrict__ ei,
    float* __restrict__ aggx, float* __restrict__ aggp) {
  const int e = blockIdx.x;
  const int f = threadIdx.x;
  const int s = ei[e];
  const int d = ei[N_EDGES + e];
  if (f < IN_F) {
    unsafeAtomicAdd(&aggx[(size_t)d * IN_F + f], x[(size_t)s * IN_F + f]);
  } else {
    const int fp = f - IN_F;
    unsafeAtomicAdd(&aggp[(size_t)d * POS_F + fp], pos[(size_t)s * POS_F + fp]);
  }
}

// ---------------------------------------------------------------------------
// Fused GIN path: GEMM1 -> LN -> ReLU -> GEMM2 -> LN -> ReLU (+residual)
// One wave computes one 16-row tile entirely with V_WMMA_F32_16X16X4_F32.
// ---------------------------------------------------------------------------
template <bool CONCAT, bool RESID>
__global__ __launch_bounds__(128) void gin_path_kernel(
    const float* __restrict__ x,    // [N,128]
    const float* __restrict__ pos,  // [N,64]
    const float* __restrict__ aggx, // [N,128]
    const float* __restrict__ aggp, // [N,64]
    const float* __restrict__ epsv,
    const float* __restrict__ W1, const float* __restrict__ b1,
    const float* __restrict__ g1, const float* __restrict__ be1,
    const float* __restrict__ W2, const float* __restrict__ b2,
    const float* __restrict__ lng, const float* __restrict__ lnb,
    float* __restrict__ out) {
  __shared__ float L[WPB][16][132];   // padded: conflict-free row reads
  __shared__ float MU[WPB][16];
  __shared__ float RS[WPB][16];

  const int lane = threadIdx.x & 31;
  const int w    = threadIdx.x >> 5;
  int wave = blockIdx.x * WPB + w;
  const bool active = wave < NTILES;
  if (!active) wave = NTILES - 1;        // clamp: harmless redundant compute
  const int row0 = wave * 16;

  const int m  = lane & 15;              // M index of this lane
  const int gq = lane >> 4;              // K-half selector (0/1)
  const int rowA = row0 + m;
  const float ep = 1.0f + epsv[0];

  // ---------------- GEMM1: [16 x K1] @ W1[K1 x 128] ----------------
  v8f acc[8] = {};
  // -- K segment from x/agg_x (CONCAT only), branch-free loops --
  if constexpr (CONCAT) {
    const v2f* __restrict__ xr = (const v2f*)&x[(size_t)rowA * IN_F];
    const v2f* __restrict__ ar = (const v2f*)&aggx[(size_t)rowA * IN_F];
    for (int kk = 0; kk < IN_F; kk += 4) {
      const int k0 = kk + 2 * gq;        // even
      v2f a = ep * xr[k0 >> 1] + ar[k0 >> 1];
#pragma unroll
      for (int t = 0; t < 8; ++t) {
        const int n = t * 16 + m;
        v2f b; b.x = W1[(size_t)k0 * EMB_F + n];
               b.y = W1[(size_t)(k0 + 1) * EMB_F + n];
        acc[t] = __builtin_amdgcn_wmma_f32_16x16x4_f32(
            false, a, false, b, (short)0, acc[t], false, false);
      }
    }
  }
  // -- K segment from pos/agg_pos --
  {
    const int kbase = CONCAT ? IN_F : 0; // offset into W1 rows
    const v2f* __restrict__ pr = (const v2f*)&pos[(size_t)rowA * POS_F];
    const v2f* __restrict__ ar = (const v2f*)&aggp[(size_t)rowA * POS_F];
    for (int kk = 0; kk < POS_F; kk += 4) {
      const int k0 = kk + 2 * gq;        // even, local to pos segment
      v2f a = ep * pr[k0 >> 1] + ar[k0 >> 1];
#pragma unroll
      for (int t = 0; t < 8; ++t) {
        const int n = t * 16 + m;
        v2f b; b.x = W1[(size_t)(kbase + k0) * EMB_F + n];
               b.y = W1[(size_t)(kbase + k0 + 1) * EMB_F + n];
        acc[t] = __builtin_amdgcn_wmma_f32_16x16x4_f32(
            false, a, false, b, (short)0, acc[t], false, false);
      }
    }
  }
  // bias + stash in wave-private LDS (C layout -> row-major tile)
#pragma unroll
  for (int t = 0; t < 8; ++t) {
    const int n = t * 16 + m;
    const float bb = b1[n];
#pragma unroll
    for (int v = 0; v < 8; ++v) L[w][v + 8 * gq][n] = acc[t][v] + bb;
  }
  __syncthreads();
  if (lane < 16) {                       // per-row LayerNorm stats (b128 reads)
    float s = 0.f, ss = 0.f;
    const v4f* row = (const v4f*)&L[w][lane][0];
    for (int c = 0; c < EMB_F / 4; ++c) {
      v4f u = row[c];
      s  += u.x + u.y + u.z + u.w;
      ss += u.x * u.x + u.y * u.y + u.z * u.z + u.w * u.w;
    }
    const float mu = s * (1.0f / EMB_F);
    const float var = ss * (1.0f / EMB_F) - mu * mu;
    MU[w][lane] = mu;
    RS[w][lane] = rsqrtf(var + LNE);
  }
  __syncthreads();

  // ---------------- GEMM2: relu(ln(h1)) @ W2[128 x 128] ----------------
  const float mu_m = MU[w][m];
  const float rs_m = RS[w][m];
  v8f acc2[8] = {};
  for (int kk = 0; kk < EMB_F; kk += 4) {
    const int k0 = kk + 2 * gq;
    v2f h  = *(const v2f*)&L[w][m][k0];          // ds_load_b64
    v2f gg = *(const v2f*)&g1[k0];
    v2f bb = *(const v2f*)&be1[k0];
    v2f a;
    a.x = fmaxf((h.x - mu_m) * rs_m * gg.x + bb.x, 0.0f);
    a.y = fmaxf((h.y - mu_m) * rs_m * gg.y + bb.y, 0.0f);
#pragma unroll
    for (int t = 0; t < 8; ++t) {
      const int n = t * 16 + m;
      v2f b; b.x = W2[(size_t)k0 * EMB_F + n];
             b.y = W2[(size_t)(k0 + 1) * EMB_F + n];
      acc2[t] = __builtin_amdgcn_wmma_f32_16x16x4_f32(
          false, a, false, b, (short)0, acc2[t], false, false);
    }
  }
  __syncthreads();                       // done reading LDS before overwrite
#pragma unroll
  for (int t = 0; t < 8; ++t) {
    const int n = t * 16 + m;
    const float bb = b2[n];
#pragma unroll
    for (int v = 0; v < 8; ++v) L[w][v + 8 * gq][n] = acc2[t][v] + bb;
  }
  __syncthreads();
  if (lane < 16) {
    float s = 0.f, ss = 0.f;
    const v4f* row = (const v4f*)&L[w][lane][0];
    for (int c = 0; c < EMB_F / 4; ++c) {
      v4f u = row[c];
      s  += u.x + u.y + u.z + u.w;
      ss += u.x * u.x + u.y * u.y + u.z * u.z + u.w * u.w;
    }
    const float mu = s * (1.0f / EMB_F);
    const float var = ss * (1.0f / EMB_F) - mu * mu;
    MU[w][lane] = mu;
    RS[w][lane] = rsqrtf(var + LNE);
  }
  __syncthreads();

  // -------- final LN + ReLU (+ residual) + b128 store --------
  // 16 rows x 128 cols; each lane handles 4 consecutive cols per row-pass:
  // rows m2 = lane/8 + 4*pass (4 rows per pass), cols c = (lane%8)*16? --
  // simpler: flat index over v4f elements: 16*32 vectors, 32 lanes -> 16 iters
  for (int vi = lane; vi < 16 * (EMB_F / 4); vi += 32) {
    const int mm = vi >> 5;              // row (EMB_F/4 == 32 vectors per row)
    const int cv = vi & 31;              // vector-column
    v4f u  = *(const v4f*)&L[w][mm][cv * 4];
    v4f gg = *(const v4f*)&lng[cv * 4];
    v4f bb = *(const v4f*)&lnb[cv * 4];
    const float mu = MU[w][mm], rs = RS[w][mm];
    v4f val = (u - mu) * rs * gg + bb;
    val.x = fmaxf(val.x, 0.0f); val.y = fmaxf(val.y, 0.0f);
    val.z = fmaxf(val.z, 0.0f); val.w = fmaxf(val.w, 0.0f);
    if constexpr (RESID)
      val += *(const v4f*)&x[(size_t)(row0 + mm) * IN_F + cv * 4];
    if (active)
      *(v4f*)&out[(size_t)(row0 + mm) * EMB_F + cv * 4] = val;
  }
}

// ---------------------------------------------------------------------------
extern "C" void kernel_launch(void* const* d_in, const int* in_sizes, int n_in,
                              void* d_out, int out_size, void* d_ws, size_t ws_size,
                              hipStream_t stream) {
  const float* x    = (const float*)d_in[0];
  const float* pos  = (const float*)d_in[1];
  const int*   ei   = (const int*)d_in[2];
  const float* eps  = (const float*)d_in[3];
  const float* W1   = (const float*)d_in[4];
  const float* b1   = (const float*)d_in[5];
  const float* g1   = (const float*)d_in[6];
  const float* be1  = (const float*)d_in[7];
  const float* W2   = (const float*)d_in[8];
  const float* b2   = (const float*)d_in[9];
  const float* lng  = (const float*)d_in[10];
  const float* lnb  = (const float*)d_in[11];
  const float* epsp = (const float*)d_in[12];
  const float* W1p  = (const float*)d_in[13];
  const float* b1p  = (const float*)d_in[14];
  const float* g1p  = (const float*)d_in[15];
  const float* be1p = (const float*)d_in[16];
  const float* W2p  = (const float*)d_in[17];
  const float* b2p  = (const float*)d_in[18];
  const float* lnpg = (const float*)d_in[19];
  const float* lnpb = (const float*)d_in[20];

  float* aggx = (float*)d_ws;                          // [N,128]
  float* aggp = aggx + (size_t)N_NODES * IN_F;         // [N,64]
  float* outh = (float*)d_out;                         // [N,128]
  float* outp = outh + (size_t)N_NODES * EMB_F;        // [N,128]

  hipMemsetAsync(d_ws, 0, (size_t)N_NODES * (IN_F + POS_F) * sizeof(float), stream);
  scatter_kernel<<<N_EDGES, 192, 0, stream>>>(x, pos, ei, aggx, aggp);

  const int blocks = (NTILES + WPB - 1) / WPB;         // 782
  gin_path_kernel<true, true><<<blocks, 128, 0, stream>>>(
      x, pos, aggx, aggp, eps, W1, b1, g1, be1, W2, b2, lng, lnb, outh);
  gin_path_kernel<false, false><<<blocks, 128, 0, stream>>>(
      x, pos, aggx, aggp, epsp, W1p, b1p, g1p, be1p, W2p, b2p, lnpg, lnpb, outp);
}